// PINN_43817256354455
// MI455X (gfx1250) — compile-verified
//
#include <hip/hip_runtime.h>
#include <hip/hip_bf16.h>

typedef float v2f __attribute__((ext_vector_type(2)));
typedef float v8f __attribute__((ext_vector_type(8)));

#define HID    20
#define NCH    13          // jet channels: val, x,y,t, xx,xy,yy,xt,yt, xxx,xxy,xyy,yyy
#define NLAYER 7
#define CS     (16*HID)    // LDS channel stride (floats)

// One wave (32 lanes) processes 16 points. Linear part of each hidden layer is
// 13 x (16x20 @ 20x20) done with V_WMMA_F32_16X16X4_F32 (K-chunks of 4, two
// N-tiles of 16). tanh-jet nonlinearity is elementwise across the 13 C tiles.
__global__ __launch_bounds__(32)
void pinn_ns_wmma(const float* __restrict__ gx, const float* __restrict__ gy,
                  const float* __restrict__ gt,
                  const float* __restrict__ W0, const float* __restrict__ b0,
                  const float* __restrict__ Wh, const float* __restrict__ bh,
                  const float* __restrict__ Wl, const float* __restrict__ bl,
                  const float* __restrict__ lam1p, const float* __restrict__ lam2p,
                  const float* __restrict__ lb,  const float* __restrict__ ub,
                  float* __restrict__ out, int N)
{
    __shared__ __align__(16) float lds[2][NCH * CS];   // double-buffered jet H
    const int lane = threadIdx.x;       // 0..31 (wave32)
    const int m16  = lane & 15;
    const int half = lane >> 4;         // 0: lanes 0-15, 1: lanes 16-31
    const int base = blockIdx.x << 4;   // 16 points per wave

    const float lbx = lb[0], lby = lb[1], lbt = lb[2];
    const float scx = 2.0f / (ub[0] - lbx);
    const float scy = 2.0f / (ub[1] - lby);
    const float sct = 2.0f / (ub[2] - lbt);

    // ---------------- layer 0 (affine in z -> higher a-derivs are zero) -----
    {
        int p = base + m16; if (p >= N) p = N - 1;
        const float zx = scx * (gx[p] - lbx) - 1.0f;
        const float zy = scy * (gy[p] - lby) - 1.0f;
        const float zt = sct * (gt[p] - lbt) - 1.0f;
        #pragma unroll
        for (int jj = 0; jj < 10; ++jj) {
            const int j  = jj + 10 * half;                 // neurons 0..19
            const float w0 = W0[j], w1 = W0[HID + j], w2 = W0[2*HID + j];
            const float a  = zx*w0 + zy*w1 + zt*w2 + b0[j];
            const float ax = scx*w0, ay = scy*w1, at = sct*w2;
            const float h  = tanhf(a);
            const float s  = 1.0f - h*h;                   // f'
            const float f2 = -2.0f * h * s;                // f''
            const float f3 = s * (4.0f*h*h - 2.0f*s);      // f'''
            float* d = &lds[0][m16*HID + j];
            d[0]     = h;
            d[CS]    = s*ax;            d[2*CS] = s*ay;        d[3*CS] = s*at;
            d[4*CS]  = f2*ax*ax;        d[5*CS] = f2*ax*ay;    d[6*CS] = f2*ay*ay;
            d[7*CS]  = f2*ax*at;        d[8*CS] = f2*ay*at;
            d[9*CS]  = f3*ax*ax*ax;     d[10*CS]= f3*ax*ax*ay;
            d[11*CS] = f3*ax*ay*ay;     d[12*CS]= f3*ay*ay*ay;
        }
    }

    // ---------------- hidden layers: WMMA f32 16x16x4 -----------------------
    int cur = 0;
    #pragma unroll 1
    for (int L = 0; L < NLAYER; ++L) {
        const float* W  = Wh + L * HID * HID;
        const float* bb = bh + L * HID;
        const float* src = lds[cur];
        float*       dst = lds[cur ^ 1];

        #pragma unroll 1
        for (int tile = 0; tile < 2; ++tile) {
            const int   n     = m16 + 16 * tile;           // output neuron
            const bool  nok   = (n < HID);
            const float nmask = nok ? 1.0f : 0.0f;         // branch-free masking
            const int   ncl   = nok ? n : 0;               // clamped (legal) column
            const float bias  = bb[ncl] * nmask;

            // Preload all 5 B-fragments unconditionally (clamped address,
            // value masked by multiply) -> no exec-divergent loads.
            v2f bf[5];
            #pragma unroll
            for (int k = 0; k < 5; ++k) {
                const int krow = 4*k + 2*half;             // B: lanes16-31 hold K=2,3
                bf[k].x = W[ krow      * HID + ncl] * nmask;
                bf[k].y = W[(krow + 1) * HID + ncl] * nmask;
            }

            v8f acc[NCH];
            #pragma unroll
            for (int c = 0; c < NCH; ++c)
                acc[c] = (v8f){0,0,0,0,0,0,0,0};
            #pragma unroll
            for (int r = 0; r < 8; ++r) acc[0][r] = bias;  // bias only on value channel

            #pragma unroll
            for (int k = 0; k < 5; ++k) {                  // K = 20 = 5 x 4
                #pragma unroll
                for (int c = 0; c < NCH; ++c) {
                    // A frag: lane m=l%16, K pair = 4k + 2*(l/16) + {0,1}
                    v2f af = *(const v2f*)&src[c*CS + m16*HID + 4*k + 2*half];
                    acc[c] = __builtin_amdgcn_wmma_f32_16x16x4_f32(
                        false, af, false, bf[k], (short)0, acc[c], false, false);
                }
            }

            // tanh-jet chain rule, elementwise across the 13 aligned C tiles
            if (nok) {
                #pragma unroll
                for (int r = 0; r < 8; ++r) {
                    const int m = r + 8 * half;            // C layout row
                    const float a   = acc[0][r];
                    const float ax  = acc[1][r],  ay  = acc[2][r],  at  = acc[3][r];
                    const float axx = acc[4][r],  axy = acc[5][r],  ayy = acc[6][r];
                    const float axt = acc[7][r],  ayt = acc[8][r];
                    const float axxx= acc[9][r],  axxy= acc[10][r];
                    const float axyy= acc[11][r], ayyy= acc[12][r];
                    const float h  = tanhf(a);
                    const float s  = 1.0f - h*h;
                    const float f2 = -2.0f * h * s;
                    const float f3 = s * (4.0f*h*h - 2.0f*s);
                    float* d = &dst[m*HID + n];
                    d[0]     = h;
                    d[CS]    = s*ax;  d[2*CS] = s*ay;  d[3*CS] = s*at;
                    d[4*CS]  = f2*ax*ax + s*axx;
                    d[5*CS]  = f2*ax*ay + s*axy;
                    d[6*CS]  = f2*ay*ay + s*ayy;
                    d[7*CS]  = f2*ax*at + s*axt;
                    d[8*CS]  = f2*ay*at + s*ayt;
                    d[9*CS]  = f3*ax*ax*ax + 3.0f*f2*ax*axx + s*axxx;
                    d[10*CS] = f3*ax*ax*ay + f2*(axx*ay + 2.0f*ax*axy) + s*axxy;
                    d[11*CS] = f3*ax*ay*ay + f2*(ayy*ax + 2.0f*axy*ay) + s*axyy;
                    d[12*CS] = f3*ay*ay*ay + 3.0f*f2*ay*ayy + s*ayyy;
                }
            }
        }
        cur ^= 1;
    }

    // ---------------- head: psi,p = H @ Wl + bl and NS residuals ------------
    if (half == 0) {
        const float* H = lds[cur];
        const int m = m16;
        float s_x=0, s_y=0, s_xx=0, s_xy=0, s_yy=0, s_xt=0, s_yt=0;
        float s_xxx=0, s_xxy=0, s_xyy=0, s_yyy=0;
        float pv=0, p_x=0, p_y=0;
        #pragma unroll
        for (int j = 0; j < HID; ++j) {
            const float wl0 = Wl[j*2 + 0], wl1 = Wl[j*2 + 1];
            const float* hp = &H[m*HID + j];
            pv    += hp[0]     * wl1;
            s_x   += hp[CS]    * wl0;   p_x += hp[CS]   * wl1;
            s_y   += hp[2*CS]  * wl0;   p_y += hp[2*CS] * wl1;
            s_xx  += hp[4*CS]  * wl0;
            s_xy  += hp[5*CS]  * wl0;
            s_yy  += hp[6*CS]  * wl0;
            s_xt  += hp[7*CS]  * wl0;
            s_yt  += hp[8*CS]  * wl0;
            s_xxx += hp[9*CS]  * wl0;
            s_xxy += hp[10*CS] * wl0;
            s_xyy += hp[11*CS] * wl0;
            s_yyy += hp[12*CS] * wl0;
        }
        pv += bl[1];
        const float u   = s_y,   v   = -s_x;
        const float u_t = s_yt,  u_x = s_xy,  u_y = s_yy;
        const float v_t = -s_xt, v_x = -s_xx, v_y = -s_xy;
        const float u_xx = s_xxy, u_yy = s_yyy;          // psi_yxx, psi_yyy
        const float v_xx = -s_xxx, v_yy = -s_xyy;        // -psi_xxx, -psi_xyy
        const float l1 = lam1p[0], l2 = lam2p[0];
        const float f_u = u_t + l1*(u*u_x + v*u_y) + p_x - l2*(u_xx + u_yy);
        const float f_v = v_t + l1*(u*v_x + v*v_y) + p_y - l2*(v_xx + v_yy);
        const int p = base + m;
        if (p < N) {
            out[p]         = u;
            out[N + p]     = v;
            out[2*N + p]   = pv;
            out[3*N + p]   = f_u;
            out[4*N + p]   = f_v;
        }
    }
}

extern "C" void kernel_launch(void* const* d_in, const int* in_sizes, int n_in,
                              void* d_out, int out_size, void* d_ws, size_t ws_size,
                              hipStream_t stream) {
    const float* x    = (const float*)d_in[0];
    const float* y    = (const float*)d_in[1];
    const float* t    = (const float*)d_in[2];
    const float* W0   = (const float*)d_in[3];
    const float* b0   = (const float*)d_in[4];
    const float* Wh   = (const float*)d_in[5];
    const float* bh   = (const float*)d_in[6];
    const float* Wl   = (const float*)d_in[7];
    const float* bl   = (const float*)d_in[8];
    const float* lam1 = (const float*)d_in[9];
    const float* lam2 = (const float*)d_in[10];
    const float* lb   = (const float*)d_in[11];
    const float* ub   = (const float*)d_in[12];
    float* out = (float*)d_out;
    const int N = in_sizes[0];
    const int blocks = (N + 15) / 16;   // one wave (32 threads) per 16 points
    pinn_ns_wmma<<<blocks, 32, 0, stream>>>(
        x, y, t, W0, b0, Wh, bh, Wl, bl, lam1, lam2, lb, ub, out, N);
}